// WignerCombiningSingleUnrolled_77189152243959
// MI455X (gfx1250) — compile-verified
//
#include <hip/hip_runtime.h>
#include <cstdint>

// ---------------------------------------------------------------------------
// Wigner CG combining, L1=L2=LAMBD=4 (9x9 x 9x9 -> 9x9), N samples.
// out[n, mu*9+mup] = sum_{p in T[mu], q in T[mup]} X1[n,m1_p,m1_q]*X2[n,m2_p,m2_q]*mult[p*61+q]
// Index structure is compile-time; only `mult` is runtime data.
// VALU-bound (3721 fma/sample); WMMA is counterproductive (1/81 scatter density).
// CDNA5 paths: async global->LDS load staging + async LDS->global store
// (ASYNCcnt), wave32 bank-conflict-free LDS layout (odd 81-dword stride).
// ---------------------------------------------------------------------------

#define TPB  64
#define SLAB (TPB * 81)   // floats per slab per block

// ---- compile-time Clebsch-Gordan pair table (61 pairs, grouped by mu) ----
struct Tab {
  int m1[61];
  int m2[61];
  int mu[61];
  int start[10];
};
constexpr Tab make_tab() {
  Tab t{};
  int p = 0;
  for (int mu = 0; mu < 9; ++mu) {
    t.start[mu] = p;
    for (int m1 = 0; m1 < 9; ++m1) {
      int m2 = mu + 4 - m1;
      if (m2 >= 0 && m2 <= 8) { t.m1[p] = m1; t.m2[p] = m2; t.mu[p] = mu; ++p; }
    }
  }
  t.start[9] = p;
  return t;
}
constexpr Tab T = make_tab();
static_assert(make_tab().start[9] == 61, "pair count must be 61");

// ---- CDNA5 async LDS<->global helpers (guarded; fall back to sync copy) ----
typedef int v4i __attribute__((ext_vector_type(4)));
typedef __attribute__((address_space(1))) int  g_i32;
typedef __attribute__((address_space(1))) v4i  g_v4i;
typedef __attribute__((address_space(3))) int  l_i32;
typedef __attribute__((address_space(3))) v4i  l_v4i;

__device__ __forceinline__ void stage_b128(void* lds_dst, const void* gsrc) {
#if __has_builtin(__builtin_amdgcn_global_load_async_to_lds_b128)
  __builtin_amdgcn_global_load_async_to_lds_b128(
      (g_v4i*)(uintptr_t)gsrc,
      (l_v4i*)(uint32_t)(uintptr_t)lds_dst, 0, 0);
#else
  *(float4*)lds_dst = *(const float4*)gsrc;
#endif
}

__device__ __forceinline__ void stage_b32(void* lds_dst, const void* gsrc) {
#if __has_builtin(__builtin_amdgcn_global_load_async_to_lds_b32)
  __builtin_amdgcn_global_load_async_to_lds_b32(
      (g_i32*)(uintptr_t)gsrc,
      (l_i32*)(uint32_t)(uintptr_t)lds_dst, 0, 0);
#else
  *(float*)lds_dst = *(const float*)gsrc;
#endif
}

__device__ __forceinline__ void drain_b128(void* gdst, const void* lds_src) {
#if __has_builtin(__builtin_amdgcn_global_store_async_from_lds_b128)
  __builtin_amdgcn_global_store_async_from_lds_b128(
      (g_v4i*)(uintptr_t)gdst,
      (l_v4i*)(uint32_t)(uintptr_t)lds_src, 0, 0);
#else
  *(float4*)gdst = *(const float4*)lds_src;
#endif
}

__device__ __forceinline__ void drain_b32(void* gdst, const void* lds_src) {
#if __has_builtin(__builtin_amdgcn_global_store_async_from_lds_b32)
  __builtin_amdgcn_global_store_async_from_lds_b32(
      (g_i32*)(uintptr_t)gdst,
      (l_i32*)(uint32_t)(uintptr_t)lds_src, 0, 0);
#else
  *(float*)gdst = *(const float*)lds_src;
#endif
}

__device__ __forceinline__ void wait_async_lds() {
  // Safe even if the fallback (sync) staging path was used: counter is 0.
  asm volatile("s_wait_asynccnt 0x0" ::: "memory");
}

// ---- unrolled accumulation for one output row mu (9 buckets) ----
// Rows are fetched from LDS per pair p: there is no intra-mu row reuse, so a
// VGPR-resident X1 cache only trades ~550 merged DS loads for catastrophic
// (>256 VGPR, MSB-switching) register pressure. Keep live set ~60 VGPRs.
template <int MU>
__device__ __forceinline__ void accum_mu(const float* __restrict__ x1base,
                                         const float* __restrict__ x2base,
                                         const float* __restrict__ mult,
                                         float* __restrict__ ldsout) {
  float acc[9] = {0.f, 0.f, 0.f, 0.f, 0.f, 0.f, 0.f, 0.f, 0.f};
#pragma unroll
  for (int p = T.start[MU]; p < T.start[MU + 1]; ++p) {
    float x1r[9], x2r[9];
#pragma unroll
    for (int j = 0; j < 9; ++j) x1r[j] = x1base[T.m1[p] * 9 + j];
#pragma unroll
    for (int j = 0; j < 9; ++j) x2r[j] = x2base[T.m2[p] * 9 + j];
    const float* mrow = mult + p * 61;  // wave-uniform -> scalar s_load path
#pragma unroll
    for (int q = 0; q < 61; ++q) {
      acc[T.mu[q]] = fmaf(x1r[T.m1[q]] * x2r[T.m2[q]], mrow[q], acc[T.mu[q]]);
    }
  }
#pragma unroll
  for (int c = 0; c < 9; ++c) ldsout[MU * 9 + c] = acc[c];
}

__global__ __launch_bounds__(TPB) void wigner_cg_kernel(
    const float* __restrict__ X1, const float* __restrict__ X2,
    const float* __restrict__ mult, float* __restrict__ out, int n_total) {
  __shared__ float lds1[SLAB];
  __shared__ float lds2[SLAB];
  __shared__ float ldso[SLAB];

  const int tid = threadIdx.x;
  const long long blk = blockIdx.x;
  const long long base = blk * (long long)SLAB;   // element offset of slab
  const long long total = (long long)n_total * 81;
  const long long remll = total - base;
  const int valid = remll < (long long)SLAB ? (int)remll : SLAB;
  const int nf4 = valid >> 2;

  const float* g1 = X1 + base;  // 16B aligned: blk*SLAB*4 = blk*20736 bytes
  const float* g2 = X2 + base;

  // Coalesced async staging: bulk b128, scalar b32 tail (generic-safe).
  for (int i = tid; i < nf4; i += TPB) {
    stage_b128(&lds1[4 * i], g1 + 4 * i);
    stage_b128(&lds2[4 * i], g2 + 4 * i);
  }
  for (int f = (nf4 << 2) + tid; f < valid; f += TPB) {
    stage_b32(&lds1[f], g1 + f);
    stage_b32(&lds2[f], g2 + f);
  }
  wait_async_lds();
  __syncthreads();

  // One sample per lane; 81-dword lane stride is odd -> conflict-free (wave32,
  // 64 banks: 81*dl == 0 mod 64 requires dl == 0 mod 32... never within a wave).
  const float* x1base = &lds1[tid * 81];
  const float* x2base = &lds2[tid * 81];
  float* myout = &ldso[tid * 81];

  accum_mu<0>(x1base, x2base, mult, myout);
  accum_mu<1>(x1base, x2base, mult, myout);
  accum_mu<2>(x1base, x2base, mult, myout);
  accum_mu<3>(x1base, x2base, mult, myout);
  accum_mu<4>(x1base, x2base, mult, myout);
  accum_mu<5>(x1base, x2base, mult, myout);
  accum_mu<6>(x1base, x2base, mult, myout);
  accum_mu<7>(x1base, x2base, mult, myout);
  accum_mu<8>(x1base, x2base, mult, myout);

  __syncthreads();

  // Coalesced async drain LDS -> global (bounded by `valid`: tail-safe, no
  // per-lane predicates needed; S_ENDPGM implicitly waits all counters).
  float* gout = out + base;
  for (int i = tid; i < nf4; i += TPB) {
    drain_b128(gout + 4 * i, &ldso[4 * i]);
  }
  for (int f = (nf4 << 2) + tid; f < valid; f += TPB) {
    drain_b32(gout + f, &ldso[f]);
  }
}

extern "C" void kernel_launch(void* const* d_in, const int* in_sizes, int n_in,
                              void* d_out, int out_size, void* d_ws, size_t ws_size,
                              hipStream_t stream) {
  // setup_inputs order: X1, X2, m1, m1p, m2, m2p, mult, mu_both
  const float* X1   = (const float*)d_in[0];
  const float* X2   = (const float*)d_in[1];
  const float* mult = (const float*)d_in[6];
  float* out = (float*)d_out;

  const int n = in_sizes[0] / 81;  // N samples (X1 is N*9*9)
  const int blocks = (n + TPB - 1) / TPB;
  wigner_cg_kernel<<<blocks, TPB, 0, stream>>>(X1, X2, mult, out, n);
}